// HeteroAttentionNet_68289980006599
// MI455X (gfx1250) — compile-verified
//
#include <hip/hip_runtime.h>
#include <math.h>

// ---------------------------------------------------------------------------
// HeteroAttentionNet (HANConv x4) for gfx1250.
//   - Projections use V_WMMA_F32_16X16X4_F32 (full f32, documented layouts).
//   - Edge softmax/aggregation via global float atomics (memory-bound part).
//   - Workspace requirement ~324 MB (see offsets in kernel_launch).
// ---------------------------------------------------------------------------

typedef __attribute__((ext_vector_type(2))) float v2f;
typedef __attribute__((ext_vector_type(8))) float v8f;

#define HID 32

// ---------------- encoder: out = relu(x @ W + b), x is [N,K] ----------------
__global__ void enc_kernel(const float* __restrict__ x, int K,
                           const float* __restrict__ W, const float* __restrict__ b,
                           float* __restrict__ out, int N) {
  long long t = (long long)blockIdx.x * blockDim.x + threadIdx.x;
  if (t >= (long long)N * HID) return;
  int c = (int)(t & (HID - 1));
  long long n = t >> 5;
  const float* xr = x + n * K;
  float acc = b[c];
  for (int k = 0; k < K; ++k) acc = fmaf(xr[k], W[k * HID + c], acc);
  out[n * HID + c] = fmaxf(acc, 0.0f);
}

// ------------- projection: out[N,32] = in[N,32] @ W[32,32] + b --------------
// One wave computes a 16(M)x32(N) tile = two 16x16 C fragments, accumulating
// with 8 chained V_WMMA_F32_16X16X4_F32 per fragment over K=32.
// A frag (16x4 f32): lanes 0-15 hold (M=lane, K=k0,k0+1), lanes 16-31 hold
// (M=lane-16, K=k0+2,k0+3). B frag (4x16) mirrors it. C frag: VGPR j holds
// (M=j + 8*(lane>=16), N=(lane&15)+16*nh).  N must be a multiple of 16.
__global__ void proj_wmma(const float* __restrict__ in, const float* __restrict__ W,
                          const float* __restrict__ bias, float* __restrict__ out, int N) {
  int wave = (int)((blockIdx.x * (long long)blockDim.x + threadIdx.x) >> 5);
  int lane = threadIdx.x & 31;
  int m0 = wave * 16;
  if (m0 >= N) return;                  // wave-uniform: EXEC stays all-ones
  int col = lane & 15;
  int hi  = lane >> 4;                  // 0: lanes 0-15, 1: lanes 16-31
  int ka  = hi * 2;                     // K offset inside each 4-wide step
  const float* arow = in + (size_t)(m0 + col) * HID;

  v8f c0, c1;
  float b0v = bias[col], b1v = bias[16 + col];
#pragma unroll
  for (int j = 0; j < 8; ++j) { c0[j] = b0v; c1[j] = b1v; }

#pragma unroll
  for (int k0 = 0; k0 < HID; k0 += 4) {
    v2f a;
    a.x = arow[k0 + ka];
    a.y = arow[k0 + ka + 1];
    v2f b0, b1;
    b0.x = W[(k0 + ka) * HID + col];
    b0.y = W[(k0 + ka + 1) * HID + col];
    b1.x = W[(k0 + ka) * HID + 16 + col];
    b1.y = W[(k0 + ka + 1) * HID + 16 + col];
    c0 = __builtin_amdgcn_wmma_f32_16x16x4_f32(false, a, false, b0, (short)0, c0, false, false);
    c1 = __builtin_amdgcn_wmma_f32_16x16x4_f32(false, a, false, b1, (short)0, c1, false, false);
  }

#pragma unroll
  for (int j = 0; j < 8; ++j) {
    size_t m = (size_t)(m0 + j + hi * 8);
    out[m * HID + col]      = c0[j];
    out[m * HID + 16 + col] = c1[j];
  }
}

// ---- per-node attention coefficients: o_v[n,h] = sum_d h[n,h,d]*v[h,d] ----
__global__ void attn_dots(const float* __restrict__ h, int N,
                          const float* v0, const float* v1,
                          const float* v2, const float* v3,
                          float* o0, float* o1, float* o2, float* o3, int nv) {
  int n = blockIdx.x * blockDim.x + threadIdx.x;
  if (n >= N) return;
  float hv[32];
  const float4* hp = (const float4*)(h + (size_t)n * HID);
#pragma unroll
  for (int i = 0; i < 8; ++i) {
    float4 t = hp[i];
    hv[4 * i + 0] = t.x; hv[4 * i + 1] = t.y; hv[4 * i + 2] = t.z; hv[4 * i + 3] = t.w;
  }
  const float* vs[4] = { v0, v1, v2, v3 };
  float* os[4] = { o0, o1, o2, o3 };
  for (int v = 0; v < nv; ++v) {
    const float* vv = vs[v];
    float* ov = os[v] + (size_t)n * 4;
#pragma unroll
    for (int hh = 0; hh < 4; ++hh) {
      float s = 0.0f;
#pragma unroll
      for (int d = 0; d < 8; ++d) s = fmaf(hv[hh * 8 + d], vv[hh * 8 + d], s);
      ov[hh] = s;
    }
  }
}

// ---- order-preserving float<->uint encoding so atomicMax(uint) == float max;
//      memset(0) acts as -inf (enc(-inf)=0x007FFFFF > 0). ----
__device__ __forceinline__ unsigned fenc(float f) {
  unsigned u = __float_as_uint(f);
  return (u & 0x80000000u) ? ~u : (u | 0x80000000u);
}
__device__ __forceinline__ float fdec(unsigned u) {
  return (u & 0x80000000u) ? __uint_as_float(u & 0x7fffffffu) : __uint_as_float(~u);
}

__device__ __forceinline__ float leaky02(float v) { return v > 0.0f ? v : 0.2f * v; }

// ---- edge pass 1: segment max of leaky_relu(asrc[si]+adst[di]) per head ----
__global__ void edge_pass1(const int* __restrict__ si, const int* __restrict__ di,
                           const float* __restrict__ asrc, const float* __restrict__ adst,
                           unsigned* __restrict__ amax, int E) {
  int e = blockIdx.x * blockDim.x + threadIdx.x;
  if (e >= E) return;
  int s = si[e], d = di[e];
  float4 a = *(const float4*)(asrc + (size_t)s * 4);
  float4 b = *(const float4*)(adst + (size_t)d * 4);
  unsigned* am = amax + (size_t)d * 4;
  atomicMax(&am[0], fenc(leaky02(a.x + b.x)));
  atomicMax(&am[1], fenc(leaky02(a.y + b.y)));
  atomicMax(&am[2], fenc(leaky02(a.z + b.z)));
  atomicMax(&am[3], fenc(leaky02(a.w + b.w)));
}

// ---- edge pass 2: ex = exp(alpha - max); store ex, accumulate denominator ----
__global__ void edge_pass2(const int* __restrict__ si, const int* __restrict__ di,
                           const float* __restrict__ asrc, const float* __restrict__ adst,
                           const unsigned* __restrict__ amax,
                           float* __restrict__ ex, float* __restrict__ den, int E) {
  int e = blockIdx.x * blockDim.x + threadIdx.x;
  if (e >= E) return;
  int s = si[e], d = di[e];
  float4 a = *(const float4*)(asrc + (size_t)s * 4);
  float4 b = *(const float4*)(adst + (size_t)d * 4);
  const unsigned* am = amax + (size_t)d * 4;
  float* dn = den + (size_t)d * 4;
  float* exo = ex + (size_t)e * 4;
  float e0 = __expf(leaky02(a.x + b.x) - fdec(am[0]));
  float e1 = __expf(leaky02(a.y + b.y) - fdec(am[1]));
  float e2 = __expf(leaky02(a.z + b.z) - fdec(am[2]));
  float e3 = __expf(leaky02(a.w + b.w) - fdec(am[3]));
  exo[0] = e0; exo[1] = e1; exo[2] = e2; exo[3] = e3;
  atomicAdd(&dn[0], e0); atomicAdd(&dn[1], e1);
  atomicAdd(&dn[2], e2); atomicAdd(&dn[3], e3);
}

// ---- edge pass 3: agg[di] += hs[si] * (ex / (den[di]+1e-16)); one thread per
//      (edge, head) -> 8 f32 atomic adds ----
__global__ void edge_pass3(const int* __restrict__ si, const int* __restrict__ di,
                           const float* __restrict__ ex, const float* __restrict__ den,
                           const float* __restrict__ hs, float* __restrict__ agg, int E) {
  long long g = (long long)blockIdx.x * blockDim.x + threadIdx.x;
  if (g >= 4LL * E) return;
  int e = (int)(g >> 2), h = (int)(g & 3);
  int s = si[e], d = di[e];
  float w = ex[(size_t)e * 4 + h] / (den[(size_t)d * 4 + h] + 1e-16f);
  const float* hr = hs + (size_t)s * HID + h * 8;
  float* ar = agg + (size_t)d * HID + h * 8;
  float4 h0 = *(const float4*)(hr);
  float4 h1 = *(const float4*)(hr + 4);
  atomicAdd(&ar[0], h0.x * w); atomicAdd(&ar[1], h0.y * w);
  atomicAdd(&ar[2], h0.z * w); atomicAdd(&ar[3], h0.w * w);
  atomicAdd(&ar[4], h1.x * w); atomicAdd(&ar[5], h1.y * w);
  atomicAdd(&ar[6], h1.z * w); atomicAdd(&ar[7], h1.w * w);
}

// ---- semantic attention partial: sumout[c] += sum_n tanh( (relu(agg[n]) @ kw + kb)[c] )
__global__ void sem_reduce(const float* __restrict__ agg, int N,
                           const float* __restrict__ kw, const float* __restrict__ kb,
                           float* __restrict__ sumout) {
  __shared__ float skw[HID * HID];
  __shared__ float skb[HID];
  __shared__ float acc[HID];
  int tl = threadIdx.x;
  for (int i = tl; i < HID * HID; i += blockDim.x) skw[i] = kw[i];
  if (tl < HID) { skb[tl] = kb[tl]; acc[tl] = 0.0f; }
  __syncthreads();
  int n = blockIdx.x * blockDim.x + tl;
  if (n < N) {
    float xv[32];
    const float4* ap = (const float4*)(agg + (size_t)n * HID);
#pragma unroll
    for (int i = 0; i < 8; ++i) {
      float4 t = ap[i];
      xv[4 * i + 0] = fmaxf(t.x, 0.0f); xv[4 * i + 1] = fmaxf(t.y, 0.0f);
      xv[4 * i + 2] = fmaxf(t.z, 0.0f); xv[4 * i + 3] = fmaxf(t.w, 0.0f);
    }
    for (int c = 0; c < HID; ++c) {
      float s = skb[c];
#pragma unroll
      for (int k = 0; k < HID; ++k) s = fmaf(xv[k], skw[k * HID + c], s);
      atomicAdd(&acc[c], tanhf(s));
    }
  }
  __syncthreads();
  if (tl < HID) atomicAdd(&sumout[tl], acc[tl]);
}

// ---- softmax over the two tx relation scores ----
__global__ void score_kernel(const float* __restrict__ sem, const float* __restrict__ q,
                             float invN, float* __restrict__ attn) {
  if (blockIdx.x == 0 && threadIdx.x == 0) {
    float s0 = 0.0f, s1 = 0.0f;
    for (int c = 0; c < HID; ++c) {
      s0 = fmaf(q[c], sem[c] * invN, s0);
      s1 = fmaf(q[c], sem[HID + c] * invN, s1);
    }
    float m = fmaxf(s0, s1);
    float e0 = __expf(s0 - m), e1 = __expf(s1 - m);
    float inv = 1.0f / (e0 + e1);
    attn[0] = e0 * inv;
    attn[1] = e1 * inv;
  }
}

__device__ __forceinline__ void layernorm_write(float* __restrict__ dst, const float* v,
                                                const float* __restrict__ g,
                                                const float* __restrict__ b) {
  float m = 0.0f;
#pragma unroll
  for (int c = 0; c < HID; ++c) m += v[c];
  m *= (1.0f / HID);
  float var = 0.0f;
#pragma unroll
  for (int c = 0; c < HID; ++c) { float dvc = v[c] - m; var = fmaf(dvc, dvc, var); }
  var *= (1.0f / HID);
  float inv = rsqrtf(var + 1e-5f);
#pragma unroll
  for (int c = 0; c < HID; ++c) dst[c] = (v[c] - m) * inv * g[c] + b[c];
}

// ---- addr update: xa = LN(relu(agg1) + xa)  (semantic group is identity) ----
__global__ void update_addr(float* __restrict__ xs, const float* __restrict__ agg,
                            const float* __restrict__ g, const float* __restrict__ b, int N) {
  int n = blockIdx.x * blockDim.x + threadIdx.x;
  if (n >= N) return;
  float v[32];
  float* xr = xs + (size_t)n * HID;
  const float* ar = agg + (size_t)n * HID;
#pragma unroll
  for (int c = 0; c < HID; ++c) v[c] = fmaxf(ar[c], 0.0f) + xr[c];
  layernorm_write(xr, v, g, b);
}

// ---- tx update: xt = LN(relu(w0*relu(agg0)+w1*relu(agg2)) + xt) ----
__global__ void update_tx(float* __restrict__ xs, const float* __restrict__ agg0,
                          const float* __restrict__ agg2, const float* __restrict__ attn,
                          const float* __restrict__ g, const float* __restrict__ b, int N) {
  int n = blockIdx.x * blockDim.x + threadIdx.x;
  if (n >= N) return;
  float w0 = attn[0], w1 = attn[1];
  float v[32];
  float* xr = xs + (size_t)n * HID;
  const float* a0 = agg0 + (size_t)n * HID;
  const float* a2 = agg2 + (size_t)n * HID;
#pragma unroll
  for (int c = 0; c < HID; ++c) {
    float ot = w0 * fmaxf(a0[c], 0.0f) + w1 * fmaxf(a2[c], 0.0f);
    v[c] = fmaxf(ot, 0.0f) + xr[c];
  }
  layernorm_write(xr, v, g, b);
}

// ---- output head: out[n,0:2] = xa[n] @ lin_w + lin_b ----
__global__ void final_kernel(const float* __restrict__ xs, const float* __restrict__ w,
                             const float* __restrict__ b, float* __restrict__ out, int N) {
  int n = blockIdx.x * blockDim.x + threadIdx.x;
  if (n >= N) return;
  const float* xr = xs + (size_t)n * HID;
  float s0 = b[0], s1 = b[1];
#pragma unroll
  for (int c = 0; c < HID; ++c) {
    float xv = xr[c];
    s0 = fmaf(xv, w[c * 2 + 0], s0);
    s1 = fmaf(xv, w[c * 2 + 1], s1);
  }
  out[(size_t)n * 2 + 0] = s0;
  out[(size_t)n * 2 + 1] = s1;
}

// ---------------------------------------------------------------------------
extern "C" void kernel_launch(void* const* d_in, const int* in_sizes, int n_in,
                              void* d_out, int out_size, void* d_ws, size_t ws_size,
                              hipStream_t stream) {
  const float* x_addr = (const float*)d_in[0];
  const float* x_tx   = (const float*)d_in[1];
  const int*   ei_in  = (const int*)d_in[2];
  const int*   ei_out = (const int*)d_in[3];
  const int*   ei_sp  = (const int*)d_in[4];
  const float* enc_w_addr = (const float*)d_in[5];
  const float* enc_b_addr = (const float*)d_in[6];
  const float* enc_w_tx   = (const float*)d_in[7];
  const float* enc_b_tx   = (const float*)d_in[8];
  const float* ln_a_g = (const float*)d_in[9];
  const float* ln_a_b = (const float*)d_in[10];
  const float* ln_t_g = (const float*)d_in[11];
  const float* ln_t_b = (const float*)d_in[12];
  const float* pw_a = (const float*)d_in[13];
  const float* pb_a = (const float*)d_in[14];
  const float* pw_t = (const float*)d_in[15];
  const float* pb_t = (const float*)d_in[16];
  const float* att_src = (const float*)d_in[17];
  const float* att_dst = (const float*)d_in[18];
  const float* klw = (const float*)d_in[19];
  const float* klb = (const float*)d_in[20];
  const float* qv  = (const float*)d_in[21];
  const float* lin_w = (const float*)d_in[22];
  const float* lin_b = (const float*)d_in[23];
  float* outp = (float*)d_out;

  const int NA  = in_sizes[0] / 53;
  const int NT  = in_sizes[1] / 6;
  const int EIN = in_sizes[2] / 2;
  const int EO  = in_sizes[3] / 2;
  const int ESP = in_sizes[4] / 2;
  const int NMAX = NA > NT ? NA : NT;
  int EMAX = EIN; if (EO > EMAX) EMAX = EO; if (ESP > EMAX) EMAX = ESP;

  // -------- workspace layout (floats); total ~81M floats (~324 MB) --------
  float* p = (float*)d_ws;
  size_t o = 0;
  auto alloc = [&](size_t nf) { float* r = p + o; o += nf; return r; };
  float* xa    = alloc((size_t)NA * HID);
  float* xt    = alloc((size_t)NT * HID);
  float* ha    = alloc((size_t)NA * HID);
  float* ht    = alloc((size_t)NT * HID);
  float* agg0  = alloc((size_t)NT * HID);   // dst = tx   (rel 0)
  float* agg1  = alloc((size_t)NA * HID);   // dst = addr (rel 1)
  float* agg2  = alloc((size_t)NT * HID);   // dst = tx   (rel 2)
  float* asrc0 = alloc((size_t)NA * 4);
  float* adst1 = alloc((size_t)NA * 4);
  float* asrc1 = alloc((size_t)NT * 4);
  float* adst0 = alloc((size_t)NT * 4);
  float* asrc2 = alloc((size_t)NT * 4);
  float* adst2 = alloc((size_t)NT * 4);
  unsigned* amax = (unsigned*)alloc((size_t)NMAX * 4);
  float* den   = alloc((size_t)NMAX * 4);
  float* ex    = alloc((size_t)EMAX * 4);
  float* sem   = alloc(64);
  float* attnw = alloc(2);
  (void)ws_size; (void)n_in; (void)out_size;

  const int B = 256;

  // -------- node encoders --------
  {
    long long ta = (long long)NA * HID, tt = (long long)NT * HID;
    enc_kernel<<<(int)((ta + B - 1) / B), B, 0, stream>>>(x_addr, 53, enc_w_addr, enc_b_addr, xa, NA);
    enc_kernel<<<(int)((tt + B - 1) / B), B, 0, stream>>>(x_tx, 6, enc_w_tx, enc_b_tx, xt, NT);
  }

  auto run_rel = [&](const int* ei, int E, const float* asrc, const float* adst,
                     const float* hs, float* agg, int Ndst) {
    hipMemsetAsync(amax, 0, (size_t)Ndst * 4 * sizeof(unsigned), stream);   // enc(-inf) bottom
    hipMemsetAsync(den,  0, (size_t)Ndst * 4 * sizeof(float), stream);
    hipMemsetAsync(agg,  0, (size_t)Ndst * HID * sizeof(float), stream);
    const int* si = ei;
    const int* di = ei + E;
    edge_pass1<<<(E + B - 1) / B, B, 0, stream>>>(si, di, asrc, adst, amax, E);
    edge_pass2<<<(E + B - 1) / B, B, 0, stream>>>(si, di, asrc, adst, amax, ex, den, E);
    long long t3 = 4LL * E;
    edge_pass3<<<(int)((t3 + B - 1) / B), B, 0, stream>>>(si, di, ex, den, hs, agg, E);
  };

  for (int l = 0; l < 4; ++l) {
    const float* PWA = pw_a + (size_t)l * HID * HID;
    const float* PBA = pb_a + (size_t)l * HID;
    const float* PWT = pw_t + (size_t)l * HID * HID;
    const float* PBT = pb_t + (size_t)l * HID;
    const float* AS  = att_src + (size_t)l * 3 * HID;   // [R][H][D] = [3][32]
    const float* AD  = att_dst + (size_t)l * 3 * HID;

    // projections via WMMA (N multiple of 16 assumed: 300000 = 18750*16)
    int tilesA = (NA + 15) / 16, tilesT = (NT + 15) / 16;
    proj_wmma<<<(tilesA + 3) / 4, 128, 0, stream>>>(xa, PWA, PBA, ha, NA);
    proj_wmma<<<(tilesT + 3) / 4, 128, 0, stream>>>(xt, PWT, PBT, ht, NT);

    // attention coefficients
    attn_dots<<<(NA + B - 1) / B, B, 0, stream>>>(ha, NA, AS + 0 * HID, AD + 1 * HID,
                                                  nullptr, nullptr,
                                                  asrc0, adst1, nullptr, nullptr, 2);
    attn_dots<<<(NT + B - 1) / B, B, 0, stream>>>(ht, NT, AS + 1 * HID, AD + 0 * HID,
                                                  AS + 2 * HID, AD + 2 * HID,
                                                  asrc1, adst0, asrc2, adst2, 4);

    // relations: (addr -input-> tx), (tx -output-> addr), (tx -spent-> tx)
    run_rel(ei_in,  EIN, asrc0, adst0, ha, agg0, NT);
    run_rel(ei_out, EO,  asrc1, adst1, ht, agg1, NA);
    run_rel(ei_sp,  ESP, asrc2, adst2, ht, agg2, NT);

    // semantic attention for tx (addr side has 1 relation -> weight == 1)
    hipMemsetAsync(sem, 0, 64 * sizeof(float), stream);
    sem_reduce<<<(NT + B - 1) / B, B, 0, stream>>>(agg0, NT, klw + (size_t)l * HID * HID,
                                                   klb + (size_t)l * HID, sem);
    sem_reduce<<<(NT + B - 1) / B, B, 0, stream>>>(agg2, NT, klw + (size_t)l * HID * HID,
                                                   klb + (size_t)l * HID, sem + HID);
    score_kernel<<<1, 32, 0, stream>>>(sem, qv + (size_t)l * HID, 1.0f / (float)NT, attnw);

    // residual + layernorm (in place; one thread owns one node row)
    update_addr<<<(NA + B - 1) / B, B, 0, stream>>>(xa, agg1, ln_a_g, ln_a_b, NA);
    update_tx<<<(NT + B - 1) / B, B, 0, stream>>>(xt, agg0, agg2, attnw, ln_t_g, ln_t_b, NT);
  }

  final_kernel<<<(NA + B - 1) / B, B, 0, stream>>>(xa, lin_w, lin_b, outp, NA);
}